// LowRankController_10359461118153
// MI455X (gfx1250) — compile-verified
//
#include <hip/hip_runtime.h>

typedef __attribute__((ext_vector_type(2))) float v2f;
typedef __attribute__((ext_vector_type(8))) float v8f;

#define DIM  4096
#define RANK 16
#define ROWS_PER_WG 16
#define WAVES 8

// out = e + (e @ B) @ A^T, fused. f32 WMMA 16x16x4, wave32.
__global__ __launch_bounds__(256) void lowrank_fused_kernel(
    const float* __restrict__ e, const float* __restrict__ A,
    const float* __restrict__ B, float* __restrict__ out)
{
    // per-wave partial t matrices (16x16 each) for cross-wave reduction
    __shared__ float lds_t[WAVES][16][16];

    const int tid  = threadIdx.x;
    const int lane = tid & 31;
    const int w    = tid >> 5;     // wave id 0..7
    const int half = lane >> 4;    // 0 = lanes 0-15, 1 = lanes 16-31
    const int lr   = lane & 15;

    const long r0 = (long)blockIdx.x * ROWS_PER_WG;   // first row of tile

    // ---------------- Stage 1: t_partial = e_tile @ B over this wave's K-slice
    // A-frag (e): lane L<16 -> row M=L, cols (k, k+1); lanes 16-31 -> (k+2, k+3)
    // B-frag (B): v0 = B[k+2*half][n], v1 = B[k+2*half+1][n], n = lane%16
    const float* erow = e + (r0 + lr) * DIM;
    v8f tacc = {};
    const int kbase = w * (DIM / WAVES);   // 512 columns per wave
    #pragma unroll 4
    for (int s = 0; s < DIM / WAVES; s += 4) {
        const int k = kbase + s;
        v2f a = *(const v2f*)(erow + k + 2 * half);
        v2f b;
        b.x = B[(k + 2 * half) * RANK + lr];
        b.y = B[(k + 2 * half + 1) * RANK + lr];
        tacc = __builtin_amdgcn_wmma_f32_16x16x4_f32(false, a, false, b,
                                                     (short)0, tacc, false, false);
    }
    // C/D layout: VGPR i holds row M = i + 8*half, col N = lane%16
    #pragma unroll
    for (int i = 0; i < 8; ++i)
        lds_t[w][i + 8 * half][lr] = tacc[i];
    __syncthreads();

    // ---------------- Build t as an A-fragment (sum the 8 wave partials)
    // A-frag: lane L -> M = L%16, per 4-chunk c4 holds K = 4*c4 + 2*half + {0,1}
    v2f afrag[4];
    #pragma unroll
    for (int c4 = 0; c4 < 4; ++c4) {
        const int k0 = 4 * c4 + 2 * half;
        float sx = 0.0f, sy = 0.0f;
        #pragma unroll
        for (int ww = 0; ww < WAVES; ++ww) {
            sx += lds_t[ww][lr][k0];
            sy += lds_t[ww][lr][k0 + 1];
        }
        afrag[c4].x = sx;
        afrag[c4].y = sy;
    }

    // ---------------- Stage 2: out_tile = e_tile + t @ A^T
    // B-frag element [k][n] = A[n0+n][k0+k]; A rows are 16 contiguous floats,
    // so each chunk is one aligned float2 load.
    for (int blk = w; blk < DIM / 16; blk += WAVES) {
        const int n0 = blk * 16;
        v8f acc;
        #pragma unroll
        for (int i = 0; i < 8; ++i)                    // C init = e (the "+ e" term)
            acc[i] = e[(r0 + i + 8 * half) * DIM + n0 + lr];
        #pragma unroll
        for (int c4 = 0; c4 < 4; ++c4) {
            v2f b = *(const v2f*)(A + (n0 + lr) * RANK + 4 * c4 + 2 * half);
            acc = __builtin_amdgcn_wmma_f32_16x16x4_f32(false, afrag[c4], false, b,
                                                        (short)0, acc, false, false);
        }
        #pragma unroll
        for (int i = 0; i < 8; ++i)
            out[(r0 + i + 8 * half) * DIM + n0 + lr] = acc[i];
    }
}

extern "C" void kernel_launch(void* const* d_in, const int* in_sizes, int n_in,
                              void* d_out, int out_size, void* d_ws, size_t ws_size,
                              hipStream_t stream) {
    const float* e = (const float*)d_in[0];   // [4, 4096, 4096] f32 (flat 16384 x 4096)
    const float* A = (const float*)d_in[1];   // [4096, 16] f32
    const float* B = (const float*)d_in[2];   // [4096, 16] f32
    float* out = (float*)d_out;               // [4, 4096, 4096] f32

    const int total_rows = 4 * DIM;           // 16384
    dim3 grid(total_rows / ROWS_PER_WG);      // 1024 workgroups
    dim3 block(256);                          // 8 wave32s
    hipLaunchKernelGGL(lowrank_fused_kernel, grid, block, 0, stream, e, A, B, out);
}